// SoftQuantize_47545287966732
// MI455X (gfx1250) — compile-verified
//
#include <hip/hip_runtime.h>

typedef float v2f __attribute__((ext_vector_type(2)));
typedef float v8f __attribute__((ext_vector_type(8)));

#define MDIM 4
#define LDIM 256
#define ROWS_PER_WAVE 16
#define WAVES_PER_BLOCK 8
#define THREADS (WAVES_PER_BLOCK * 32)
#define SIGMA_EPS 1e-4f
#define LOG2E 1.4426950408889634f

// One wave = 16 rows of z (16x4 A tile). 16 iterations of
// v_wmma_f32_16x16x4_f32 against 4x16 tiles of c give all 256 dot products.
// Softmax numerator/denominator accumulate in registers in the log2 domain
// (d2 >= 0, sig > 0 -> exp2 args <= 0, no overflow); epilogue reduces the
// 16-lane spread per row through an LDS transpose instead of a shuffle
// butterfly, then one predicated b128 store per row.
__launch_bounds__(THREADS)
__global__ void SoftQuantize_kernel(const float* __restrict__ x,
                                    const float* __restrict__ c,
                                    const float* __restrict__ sigma,
                                    float* __restrict__ out,
                                    int nrows) {
    __shared__ float4 cT_lds[LDIM];                     // cT[l] = c[0..3][l]
    __shared__ float  cn_lds[LDIM];                     // |c[:,l]|^2
    __shared__ float4 red4[WAVES_PER_BLOCK][8][32];     // o0..o3 partials
    __shared__ float  red1[WAVES_PER_BLOCK][8][32];     // sum-e partials

    const int tid = threadIdx.x;

    // Cooperative LDS fill: 256 threads, one center column each (column-major).
    {
        const int l = tid;
        float c0 = c[0 * LDIM + l];
        float c1 = c[1 * LDIM + l];
        float c2 = c[2 * LDIM + l];
        float c3 = c[3 * LDIM + l];
        cT_lds[l] = make_float4(c0, c1, c2, c3);
        cn_lds[l] = c0 * c0 + c1 * c1 + c2 * c2 + c3 * c3;
    }
    __syncthreads();

    const float sv  = sigma[0];
    const float sig = (sv > 0.0f ? sv : 0.0f) + SIGMA_EPS;
    const float nl  = -sig * LOG2E;      // exp(-sig*d2) == exp2(nl*d2)
    const float p2  = -2.0f * nl;        // coefficient of the dot product

    const int wave = tid >> 5;
    const int lane = tid & 31;
    const int half = lane >> 4;          // 0: K=0,1 & D rows 0-7 ; 1: K=2,3 & rows 8-15
    const int lm   = lane & 15;

    const int tile = blockIdx.x * WAVES_PER_BLOCK + wave;
    const int r0   = tile * ROWS_PER_WAVE;

    // ---- A tile: lane holds z[row = lm][K = 2*half .. 2*half+1]  (b64 load)
    v2f A;
    {
        const float* p = x + (size_t)(r0 + lm) * MDIM + half * 2;
        A.x = p[0];
        A.y = p[1];
    }

    // Row norms |z|^2: xor-16 pairs the two K-halves of each row.
    float rn_part = A.x * A.x + A.y * A.y;
    float rn = rn_part + __shfl_xor(rn_part, 16, 32);   // lane L: |z[L&15]|^2

    // Accumulator v needs (nl * rownorm) of row (v + 8*half), broadcast.
    float rb2[8];
#pragma unroll
    for (int v = 0; v < 8; ++v)
        rb2[v] = nl * __shfl(rn, v + 8 * half, 32);

    float se[8], o0[8], o1[8], o2[8], o3[8];
#pragma unroll
    for (int v = 0; v < 8; ++v) {
        se[v] = 0.0f;
        o0[v] = o1[v] = o2[v] = o3[v] = 0.0f;
    }

    // Limit unroll so only a few 8-VGPR WMMA accumulators are live at once
    // (full unroll pushed the kernel past VGPR 255 -> s_set_vgpr_msb traffic).
#pragma unroll 4
    for (int g = 0; g < 16; ++g) {
        const int col = g * 16 + lm;

        const float4 cc = cT_lds[col];            // one ds_load_b128
        const float cnl = nl * cn_lds[col];       // one ds_load_b32

        // B tile 4x16: lane holds c[K = 2*half + j][col] in VGPR j.
        v2f B;
        B.x = half ? cc.z : cc.x;
        B.y = half ? cc.w : cc.y;

        v8f acc = {};
        acc = __builtin_amdgcn_wmma_f32_16x16x4_f32(
            /*neg_a=*/false, A, /*neg_b=*/false, B,
            /*c_mod=*/(short)0, acc, /*reuse_a=*/false, /*reuse_b=*/false);

#pragma unroll
        for (int v = 0; v < 8; ++v) {
            // exp2(nl*(|z|^2 + |c|^2) - 2*nl*dot) : 1 add + 1 fma + v_exp
            float a = fmaf(p2, acc[v], rb2[v] + cnl);
            float e = __builtin_amdgcn_exp2f(a);
            se[v] += e;
            o0[v] = fmaf(e, cc.x, o0[v]);
            o1[v] = fmaf(e, cc.y, o1[v]);
            o2[v] = fmaf(e, cc.z, o2[v]);
            o3[v] = fmaf(e, cc.w, o3[v]);
        }
    }

    // ---- Epilogue: per-row totals. Partials for row (v + 8*half) live across
    // the 16 lanes of that half; transpose through LDS (37 ds ops vs 160
    // bpermutes for a full butterfly).
#pragma unroll
    for (int v = 0; v < 8; ++v) {
        red4[wave][v][lane] = make_float4(o0[v], o1[v], o2[v], o3[v]);
        red1[wave][v][lane] = se[v];
    }
    __syncthreads();

    // Lane pair (lm, lm+16) reduces row lm: each sums 8 of the 16 writers.
    const int hw = lm >> 3;        // which half wrote this row's partials
    const int vv = lm & 7;         // accumulator index within that half
    float4 s4 = make_float4(0.f, 0.f, 0.f, 0.f);
    float  s1 = 0.f;
#pragma unroll
    for (int j = 0; j < 8; ++j) {
        const int w = hw * 16 + half * 8 + j;
        float4 t = red4[wave][vv][w];
        s4.x += t.x; s4.y += t.y; s4.z += t.z; s4.w += t.w;
        s1   += red1[wave][vv][w];
    }
    // Combine the two 8-writer halves.
    s4.x += __shfl_xor(s4.x, 16, 32);
    s4.y += __shfl_xor(s4.y, 16, 32);
    s4.z += __shfl_xor(s4.z, 16, 32);
    s4.w += __shfl_xor(s4.w, 16, 32);
    s1   += __shfl_xor(s1,   16, 32);

    if (half == 0) {
        const int row = r0 + lm;
        const float inv = 1.0f / s1;
        float4 res = make_float4(s4.x * inv, s4.y * inv, s4.z * inv, s4.w * inv);
        *(float4*)(out + (size_t)row * MDIM) = res;     // global_store_b128
    }
}

extern "C" void kernel_launch(void* const* d_in, const int* in_sizes, int n_in,
                              void* d_out, int out_size, void* d_ws, size_t ws_size,
                              hipStream_t stream) {
    const float* x     = (const float*)d_in[0];   // (2048, 512) f32
    const float* c     = (const float*)d_in[1];   // (4, 256)   f32
    const float* sigma = (const float*)d_in[2];   // (1,)       f32
    float* out = (float*)d_out;                   // (2048, 512) f32

    const int nrows = in_sizes[0] / MDIM;                 // 262144
    const int tiles = nrows / ROWS_PER_WAVE;              // 16384 (exact)
    const int blocks = tiles / WAVES_PER_BLOCK;           // 2048  (exact)

    SoftQuantize_kernel<<<blocks, THREADS, 0, stream>>>(x, c, sigma, out, nrows);
}